// JEDIPTv3Block_58815282151830
// MI455X (gfx1250) — compile-verified
//
#include <hip/hip_runtime.h>

// Problem constants (match reference)
#define BB       32
#define NN       256
#define CC       256
#define DFFN     1024
#define GRID_DIM 128
#define LN_EPS   1e-6f
#define BN_EPS   1e-3f

typedef float v2f __attribute__((ext_vector_type(2)));
typedef float v8f __attribute__((ext_vector_type(8)));

// ---------------------------------------------------------------------------
// K1: per-batch eta/phi min -> integer grid cells (truncate, clip to [0,127])
// ---------------------------------------------------------------------------
__global__ void k_cells(const float* __restrict__ coords, int* __restrict__ cells) {
    const int b = blockIdx.x;
    const int n = threadIdx.x;                 // 256 threads
    __shared__ float se[NN], sp[NN];
    const float eta = coords[(b * NN + n) * 2 + 0];
    const float phi = coords[(b * NN + n) * 2 + 1];
    se[n] = eta; sp[n] = phi;
    __syncthreads();
    for (int off = NN / 2; off > 0; off >>= 1) {
        if (n < off) {
            se[n] = fminf(se[n], se[n + off]);
            sp[n] = fminf(sp[n], sp[n + off]);
        }
        __syncthreads();
    }
    const float emin = se[0], pmin = sp[0];
    int ge = (int)((eta - emin) / 0.1f);       // trunc toward zero (arg >= 0)
    int gp = (int)((phi - pmin) / 0.1f);
    ge = min(max(ge, 0), GRID_DIM - 1);
    gp = min(max(gp, 0), GRID_DIM - 1);
    cells[(b * NN + n) * 2 + 0] = ge;
    cells[(b * NN + n) * 2 + 1] = gp;
}

// ---------------------------------------------------------------------------
// K2: sparse depthwise conv via pairwise formulation.
// out[b,i,c] = dw_bias[c] + sum_j w[ge_j-ge_i+3, gp_j-gp_i+3, c] * x[b,j,c]
//              for deltas inside the 8x8 footprint (SAME pad lo=3, hi=4).
// One block per (b,i); 256 threads over channels.
// ---------------------------------------------------------------------------
__global__ void k_cpe(const float* __restrict__ x, const int* __restrict__ cells,
                      const float* __restrict__ dw_kernel, const float* __restrict__ dw_bias,
                      float* __restrict__ cpe) {
    const int b = blockIdx.x / NN;
    const int i = blockIdx.x % NN;
    const int c = threadIdx.x;                 // 256 threads
    __shared__ int sge[NN], sgp[NN];
    sge[c] = cells[(b * NN + c) * 2 + 0];
    sgp[c] = cells[(b * NN + c) * 2 + 1];
    __syncthreads();
    const int gei = sge[i], gpi = sgp[i];
    float acc = dw_bias[c];
    for (int j = 0; j < NN; ++j) {
        const int kh = sge[j] - gei + 3;       // uniform across block
        const int kw = sgp[j] - gpi + 3;
        if (((unsigned)kh < 8u) && ((unsigned)kw < 8u)) {
            acc += dw_kernel[(kh * 8 + kw) * CC + c] * x[((size_t)(b * NN + j)) * CC + c];
        }
    }
    cpe[((size_t)(b * NN + i)) * CC + c] = acc;
}

// ---------------------------------------------------------------------------
// f32 WMMA strip routine: one wave computes a 16-row x (NT*16)-col strip.
// A row-major [M,K], lda == K; B row-major [K,Ncols], leading dim ldb.
// A-frag (ISA 7.12.2): lanes 0-15 = rows, v0/v1 = K{0,1}; lanes 16-31 = K{2,3}.
// B-frag analogous with N striped across lanes. A-frag reused across NT wmmas.
// ---------------------------------------------------------------------------
template <int NT>
__device__ __forceinline__ void wmma_strip_f32(const float* __restrict__ A,
                                               const float* __restrict__ Bm,
                                               int row0, int col0, int K, int ldb,
                                               v8f* acc) {
    const int lane = threadIdx.x & 31;
    const int half = lane >> 4;                // 0 or 1
    const int l    = lane & 15;
    const float* arow  = A  + (size_t)(row0 + l) * K + 2 * half;
    const float* bbase = Bm + (size_t)(2 * half) * ldb + (col0 + l);
    for (int k = 0; k < K; k += 4) {
        v2f a;
        a.x = arow[k];
        a.y = arow[k + 1];
#pragma unroll
        for (int t = 0; t < NT; ++t) {
            v2f bf;
            bf.x = bbase[(size_t)k * ldb + t * 16];
            bf.y = bbase[(size_t)(k + 1) * ldb + t * 16];
            acc[t] = __builtin_amdgcn_wmma_f32_16x16x4_f32(false, a, false, bf,
                                                           (short)0, acc[t], false, false);
        }
    }
}

#define STRIP_SETUP(NCOLS)                                                    \
    const int wave = threadIdx.x >> 5;                                        \
    const int wid  = blockIdx.x * (blockDim.x >> 5) + wave;                   \
    const int groupsN = (NCOLS) / 64;                                         \
    const int row0 = (wid / groupsN) * 16;                                    \
    const int col0 = (wid % groupsN) * 64;                                    \
    const int lane = threadIdx.x & 31, half = lane >> 4, l = lane & 15;

// ---------------------------------------------------------------------------
// K3: pwout = cpe @ pw_w + pw_b       (M=8192, K=256, N=256)
// ---------------------------------------------------------------------------
__global__ void k_gemm_pw(const float* __restrict__ Amat, const float* __restrict__ W,
                          const float* __restrict__ bias, float* __restrict__ out) {
    STRIP_SETUP(CC);
    v8f acc[4];
#pragma unroll
    for (int t = 0; t < 4; ++t) {
        const float bc = bias[col0 + t * 16 + l];
#pragma unroll
        for (int e = 0; e < 8; ++e) acc[t][e] = bc;
    }
    wmma_strip_f32<4>(Amat, W, row0, col0, CC, CC, acc);
#pragma unroll
    for (int t = 0; t < 4; ++t) {
        const int col = col0 + t * 16 + l;
#pragma unroll
        for (int e = 0; e < 8; ++e) {
            const int row = row0 + e + 8 * half;
            out[(size_t)row * CC + col] = acc[t][e];
        }
    }
}

// ---------------------------------------------------------------------------
// K4: x1 = x + LayerNorm(pwout) * g + b    (one block per row of 8192)
// ---------------------------------------------------------------------------
__global__ void k_ln_res(const float* __restrict__ xin, const float* __restrict__ pwout,
                         const float* __restrict__ g, const float* __restrict__ bta,
                         float* __restrict__ x1) {
    const int row = blockIdx.x;
    const int c = threadIdx.x;                 // 256 threads
    __shared__ float red[CC];
    const float v = pwout[(size_t)row * CC + c];
    red[c] = v; __syncthreads();
    for (int off = CC / 2; off > 0; off >>= 1) {
        if (c < off) red[c] += red[c + off];
        __syncthreads();
    }
    const float mu = red[0] * (1.0f / CC);
    __syncthreads();
    const float d = v - mu;
    red[c] = d * d; __syncthreads();
    for (int off = CC / 2; off > 0; off >>= 1) {
        if (c < off) red[c] += red[c + off];
        __syncthreads();
    }
    const float var = red[0] * (1.0f / CC);
    const float rs = rsqrtf(var + LN_EPS);
    x1[(size_t)row * CC + c] = xin[(size_t)row * CC + c] + d * rs * g[c] + bta[c];
}

// ---------------------------------------------------------------------------
// K5: gvec[b] = mean_n(x1[b]) @ gi_w1 + gi_b1    (one block per batch)
// ---------------------------------------------------------------------------
__global__ void k_gctx(const float* __restrict__ x1, const float* __restrict__ w1,
                       const float* __restrict__ b1, float* __restrict__ gvec) {
    const int b = blockIdx.x, c = threadIdx.x; // 256 threads
    __shared__ float mean[CC];
    float s = 0.f;
    for (int n = 0; n < NN; ++n) s += x1[((size_t)b * NN + n) * CC + c];
    mean[c] = s * (1.0f / NN);
    __syncthreads();
    float acc = b1[c];
    for (int k = 0; k < CC; ++k) acc += mean[k] * w1[(size_t)k * CC + c];
    gvec[b * CC + c] = acc;
}

// ---------------------------------------------------------------------------
// K6: x2 = BN1( x1 + BN0( x1@gi_w2 + gi_b2 + gvec[b] ) )   (M=8192,K=256,N=256)
// ---------------------------------------------------------------------------
__global__ void k_gemm_gi(const float* __restrict__ x1, const float* __restrict__ W2,
                          const float* __restrict__ b2, const float* __restrict__ gvec,
                          const float* __restrict__ g0, const float* __restrict__ be0,
                          const float* __restrict__ m0, const float* __restrict__ v0,
                          const float* __restrict__ g1, const float* __restrict__ be1,
                          const float* __restrict__ m1, const float* __restrict__ v1,
                          float* __restrict__ x2) {
    STRIP_SETUP(CC);
    v8f acc[4];
#pragma unroll
    for (int t = 0; t < 4; ++t) {
        const float bc = b2[col0 + t * 16 + l];
#pragma unroll
        for (int e = 0; e < 8; ++e) acc[t][e] = bc;
    }
    wmma_strip_f32<4>(x1, W2, row0, col0, CC, CC, acc);
    const int bidx = row0 / NN;                // 16 | 256 -> whole strip same batch
#pragma unroll
    for (int t = 0; t < 4; ++t) {
        const int col = col0 + t * 16 + l;
        const float gv = gvec[bidx * CC + col];
        const float s0 = g0[col] * rsqrtf(v0[col] + BN_EPS);
        const float t0 = be0[col] - m0[col] * s0;
        const float s1 = g1[col] * rsqrtf(v1[col] + BN_EPS);
        const float t1 = be1[col] - m1[col] * s1;
#pragma unroll
        for (int e = 0; e < 8; ++e) {
            const int row = row0 + e + 8 * half;
            const float y  = (acc[t][e] + gv) * s0 + t0;
            const float tt = x1[(size_t)row * CC + col] + y;
            x2[(size_t)row * CC + col] = tt * s1 + t1;
        }
    }
}

// ---------------------------------------------------------------------------
// K7: h = relu(x2 @ w_ff1 + b_ff1)          (M=8192, K=256, N=1024)
// ---------------------------------------------------------------------------
__global__ void k_gemm_ffn1(const float* __restrict__ x2, const float* __restrict__ W,
                            const float* __restrict__ bias, float* __restrict__ h) {
    STRIP_SETUP(DFFN);
    v8f acc[4];
#pragma unroll
    for (int t = 0; t < 4; ++t) {
        const float bc = bias[col0 + t * 16 + l];
#pragma unroll
        for (int e = 0; e < 8; ++e) acc[t][e] = bc;
    }
    wmma_strip_f32<4>(x2, W, row0, col0, CC, DFFN, acc);
#pragma unroll
    for (int t = 0; t < 4; ++t) {
        const int col = col0 + t * 16 + l;
#pragma unroll
        for (int e = 0; e < 8; ++e) {
            const int row = row0 + e + 8 * half;
            h[(size_t)row * DFFN + col] = fmaxf(acc[t][e], 0.0f);
        }
    }
}

// ---------------------------------------------------------------------------
// K8: out = BN2( x2 + (h @ w_ff2 + b_ff2) )  (M=8192, K=1024, N=256) -> d_out
// ---------------------------------------------------------------------------
__global__ void k_gemm_ffn2(const float* __restrict__ h, const float* __restrict__ W,
                            const float* __restrict__ bias, const float* __restrict__ x2,
                            const float* __restrict__ g2, const float* __restrict__ be2,
                            const float* __restrict__ m2, const float* __restrict__ v2,
                            float* __restrict__ out) {
    STRIP_SETUP(CC);
    v8f acc[4];
#pragma unroll
    for (int t = 0; t < 4; ++t) {
        const float bc = bias[col0 + t * 16 + l];
#pragma unroll
        for (int e = 0; e < 8; ++e) acc[t][e] = bc;
    }
    wmma_strip_f32<4>(h, W, row0, col0, DFFN, CC, acc);
#pragma unroll
    for (int t = 0; t < 4; ++t) {
        const int col = col0 + t * 16 + l;
        const float s2 = g2[col] * rsqrtf(v2[col] + BN_EPS);
        const float t2 = be2[col] - m2[col] * s2;
#pragma unroll
        for (int e = 0; e < 8; ++e) {
            const int row = row0 + e + 8 * half;
            const float tt = x2[(size_t)row * CC + col] + acc[t][e];
            out[(size_t)row * CC + col] = tt * s2 + t2;
        }
    }
}

// ---------------------------------------------------------------------------
// Launcher
// ---------------------------------------------------------------------------
extern "C" void kernel_launch(void* const* d_in, const int* in_sizes, int n_in,
                              void* d_out, int out_size, void* d_ws, size_t ws_size,
                              hipStream_t stream) {
    // setup_inputs() dict order:
    const float* x         = (const float*)d_in[0];
    const float* coords    = (const float*)d_in[1];
    const float* dw_kernel = (const float*)d_in[2];
    const float* dw_bias   = (const float*)d_in[3];
    const float* pw_w      = (const float*)d_in[4];
    const float* pw_b      = (const float*)d_in[5];
    const float* ln_g      = (const float*)d_in[6];
    const float* ln_b      = (const float*)d_in[7];
    const float* gi_w1     = (const float*)d_in[8];
    const float* gi_b1     = (const float*)d_in[9];
    const float* gi_w2     = (const float*)d_in[10];
    const float* gi_b2     = (const float*)d_in[11];
    const float* bn0_g = (const float*)d_in[12]; const float* bn0_b = (const float*)d_in[13];
    const float* bn0_m = (const float*)d_in[14]; const float* bn0_v = (const float*)d_in[15];
    const float* bn1_g = (const float*)d_in[16]; const float* bn1_b = (const float*)d_in[17];
    const float* bn1_m = (const float*)d_in[18]; const float* bn1_v = (const float*)d_in[19];
    const float* bn2_g = (const float*)d_in[20]; const float* bn2_b = (const float*)d_in[21];
    const float* bn2_m = (const float*)d_in[22]; const float* bn2_v = (const float*)d_in[23];
    const float* w_ff1 = (const float*)d_in[24]; const float* b_ff1 = (const float*)d_in[25];
    const float* w_ff2 = (const float*)d_in[26]; const float* b_ff2 = (const float*)d_in[27];

    float* out = (float*)d_out;
    char*  ws  = (char*)d_ws;

    // Workspace layout (h reuses the dead pwout/x1 region; x2 stays live for K8)
    const size_t ROWS = (size_t)BB * NN;                 // 8192
    int*   cells = (int*)(ws + 0);                       //  64 KB
    float* gvec  = (float*)(ws + 65536);                 //  32 KB
    float* bufA  = (float*)(ws + 131072);                //   8 MB: cpe, later x2
    float* bufB  = (float*)(ws + 131072 + 8388608);      //   8 MB: pwout; later start of h
    float* bufC  = (float*)(ws + 131072 + 2 * 8388608);  //   8 MB: x1
    float* cpe   = bufA;
    float* pwout = bufB;
    float* x1    = bufC;
    float* x2    = bufA;                                 // cpe dead after K3
    float* h     = bufB;                                 // 32 MB span; pwout/x1 dead by K7

    // Wave = one 16x64 strip; 4 waves (128 threads) per block.
    const int stripsCC  = (int)(ROWS / 16) * (CC / 64);    // 2048 waves
    const int stripsFF1 = (int)(ROWS / 16) * (DFFN / 64);  // 8192 waves
    const int WPB = 4;

    k_cells<<<BB, NN, 0, stream>>>(coords, cells);
    k_cpe<<<BB * NN, CC, 0, stream>>>(x, cells, dw_kernel, dw_bias, cpe);
    k_gemm_pw<<<stripsCC / WPB, WPB * 32, 0, stream>>>(cpe, pw_w, pw_b, pwout);
    k_ln_res<<<(int)ROWS, CC, 0, stream>>>(x, pwout, ln_g, ln_b, x1);
    k_gctx<<<BB, CC, 0, stream>>>(x1, gi_w1, gi_b1, gvec);
    k_gemm_gi<<<stripsCC / WPB, WPB * 32, 0, stream>>>(x1, gi_w2, gi_b2, gvec,
                                                       bn0_g, bn0_b, bn0_m, bn0_v,
                                                       bn1_g, bn1_b, bn1_m, bn1_v, x2);
    k_gemm_ffn1<<<stripsFF1 / WPB, WPB * 32, 0, stream>>>(x2, w_ff1, b_ff1, h);
    k_gemm_ffn2<<<stripsCC / WPB, WPB * 32, 0, stream>>>(h, w_ff2, b_ff2, x2,
                                                         bn2_g, bn2_b, bn2_m, bn2_v, out);

    // Output tuple tail: coords passthrough ([B,N,2] floats after [B,N,C])
    hipMemcpyAsync(out + ROWS * CC, coords, (size_t)BB * NN * 2 * sizeof(float),
                   hipMemcpyDeviceToDevice, stream);
}